// CrossAttentionConditionInjection_86406152061037
// MI455X (gfx1250) — compile-verified
//
#include <hip/hip_runtime.h>

// ---------------------------------------------------------------------------
// CrossAttentionConditionInjection, MI455X (gfx1250, wave32).
//
// K/V come from `condition` broadcast over the sequence, so every key is
// identical -> softmax is exactly uniform -> attn[b,s,:] == v[b,:]. The whole
// network collapses to:
//     out[b,s,:] = ((condition[b] @ Wv^T + bv) @ Wo^T + bo)   (broadcast on s)
// Q/K projections and the 34-GFLOP attention cancel algebraically.
//
// Remaining cost: read Wv+Wo (33.5 MB) + write out (32 MB) ~ 66 MB -> ~2.8us
// at 23.3 TB/s. The two (4 x 2048) @ (2048 x 2048)^T projections are done in
// fp32 with V_WMMA_F32_16X16X4_F32 (batch rows padded to an M=16 tile; padded
// rows replicate rows 0..3 and are never stored, so EXEC stays all-1s).
// ---------------------------------------------------------------------------

typedef float v2f __attribute__((ext_vector_type(2)));
typedef float v4f __attribute__((ext_vector_type(4)));
typedef float v8f __attribute__((ext_vector_type(8)));

#define BATCH 4
#define SEQ   1024
#define DIM   2048

// Y(4,DIM) = X(4,DIM) @ W(DIM,DIM)^T + bias
// One wave per 16 output features; block = 4 waves (64 features); grid = 32.
// A-fragment (16x4 f32): lane l -> row M=l&15, cols k0+2*(l>>4) .. +1  (float2 from LDS)
// B-fragment (4x16 f32): lane l -> col N=l&15, rows k0+2*(l>>4) .. +1; B[k][n]=W[n][k]
//                        -> contiguous float2 inside W row n0+(l&15).
__global__ __launch_bounds__(128) void proj4xD_wmma(const float* __restrict__ X,
                                                    const float* __restrict__ W,
                                                    const float* __restrict__ bias,
                                                    float* __restrict__ Y) {
    __shared__ float sX[BATCH * DIM];  // 32 KB activation stage

    const int tid = threadIdx.x;
    // cooperative stage X -> LDS (2048 float4 / 128 threads = 16 each)
    const v4f* __restrict__ Xv = (const v4f*)X;
    v4f* sXv = (v4f*)sX;
#pragma unroll
    for (int i = 0; i < (BATCH * DIM / 4) / 128; ++i)
        sXv[tid + i * 128] = Xv[tid + i * 128];
    __syncthreads();

    const int lane = tid & 31;
    const int wave = tid >> 5;
    const int n0   = (blockIdx.x * 4 + wave) * 16;   // 16 output features per wave
    const int m    = lane & 15;                      // tile row (M) / tile col (N)
    const int koff = (lane >> 4) << 1;               // 0 or 2: K sub-offset per half-wave

    const float* __restrict__ Ap = &sX[(m & 3) * DIM + koff];          // rows 4..15 mirror 0..3
    const float* __restrict__ Bp = &W[(size_t)(n0 + m) * DIM + koff];  // stream one W row

    v8f c = {};  // 16x16 f32 accumulator tile (8 VGPRs)
    for (int k0 = 0; k0 < DIM; k0 += 64) {
        __builtin_prefetch(Bp + k0 + 256, 0, 0);  // global_prefetch_b8, 1KB ahead in-row
#pragma unroll
        for (int kk = 0; kk < 64; kk += 4) {
            v2f a = *(const v2f*)(Ap + k0 + kk);   // ds_load_b64
            v2f b = *(const v2f*)(Bp + k0 + kk);   // global_load_b64
            // D = A(16x4) x B(4x16) + C, fp32 end to end
            c = __builtin_amdgcn_wmma_f32_16x16x4_f32(false, a, false, b,
                                                      (short)0, c, false, false);
        }
    }

    // D layout: lane<16 holds rows M=0..7 in c[0..7]; we need rows 0..3 only.
    if (lane < 16) {
        const float bb = bias[n0 + lane];
        Y[0 * DIM + n0 + lane] = c[0] + bb;
        Y[1 * DIM + n0 + lane] = c[1] + bb;
        Y[2 * DIM + n0 + lane] = c[2] + bb;
        Y[3 * DIM + n0 + lane] = c[3] + bb;
    }
}

// out(B,S,D) = O(b,:) broadcast over s. 32 MB of coalesced float4 stores;
// O is 32 KB and stays L2-hot.
__global__ __launch_bounds__(256) void bcast_rows(const float* __restrict__ O,
                                                  float* __restrict__ out) {
    const size_t gid = (size_t)blockIdx.x * 256 + threadIdx.x;  // float4 index
    const int    dv  = (int)(gid & (DIM / 4 - 1));              // 0..511
    const size_t row = gid >> 9;                                // b*SEQ + s
    const int    b   = (int)(row >> 10);                        // row / SEQ
    ((v4f*)out)[gid] = ((const v4f*)O)[b * (DIM / 4) + dv];
}

extern "C" void kernel_launch(void* const* d_in, const int* in_sizes, int n_in,
                              void* d_out, int out_size, void* d_ws, size_t ws_size,
                              hipStream_t stream) {
    // inputs: 0 hidden_states, 1 condition, 2 Wq, 3 bq, 4 Wk, 5 bk,
    //         6 Wv, 7 bv, 8 Wo, 9 bo      (0,2..5 cancel algebraically)
    const float* condition = (const float*)d_in[1];
    const float* Wv = (const float*)d_in[6];
    const float* bv = (const float*)d_in[7];
    const float* Wo = (const float*)d_in[8];
    const float* bo = (const float*)d_in[9];

    float* V = (float*)d_ws;        // (4, DIM) = 32 KB
    float* O = V + BATCH * DIM;     // (4, DIM) = 32 KB

    dim3 blk(128), grd(DIM / 64);   // 4 waves/block * 16 cols = 64 cols/block
    proj4xD_wmma<<<grd, blk, 0, stream>>>(condition, Wv, bv, V);  // V = cond@Wv^T+bv
    proj4xD_wmma<<<grd, blk, 0, stream>>>(V, Wo, bo, O);          // O = V@Wo^T+bo

    const size_t nvec4 = (size_t)BATCH * SEQ * (DIM / 4);         // 2,097,152
    bcast_rows<<<dim3((unsigned)(nvec4 / 256)), dim3(256), 0, stream>>>(O, (float*)d_out);
}